// SAGE_L_8564164788539
// MI455X (gfx1250) — compile-verified
//
#include <hip/hip_runtime.h>

typedef __attribute__((ext_vector_type(2))) float v2f;
typedef __attribute__((ext_vector_type(8))) float v8f;

#define N_GRAPHS 512
#define BN_EPS 1e-5f

// ---------------------------------------------------------------- utilities
__global__ void zero_kernel(float* __restrict__ p, long n) {
  long i = (long)blockIdx.x * blockDim.x + threadIdx.x;
  long stride = (long)gridDim.x * blockDim.x;
  for (; i < n; i += stride) p[i] = 0.0f;
}

__global__ void degree_kernel(const int* __restrict__ dst, long E, float* __restrict__ cnt) {
  long e = (long)blockIdx.x * blockDim.x + threadIdx.x;
  if (e < E) atomicAdd(&cnt[dst[e]], 1.0f);
}

// scatter-add F-dim features over edges (float4 chunks, chunk fastest => coalesced)
// CHUNKS = F/4 is compile-time so edge/chunk split is a shift, not a division.
template <int CHUNKS>
__global__ void scatter_add_kernel(const float* __restrict__ feat,
                                   const int* __restrict__ src,
                                   const int* __restrict__ dst,
                                   long E, float* __restrict__ agg) {
  constexpr int F = CHUNKS * 4;
  long t = (long)blockIdx.x * blockDim.x + threadIdx.x;
  if (t >= E * CHUNKS) return;
  long e = t / CHUNKS;                      // CHUNKS is power of two -> shift
  int c = (int)(t - e * CHUNKS) << 2;
  int s = src[e], d = dst[e];
  const float4 v = *(const float4*)(feat + (size_t)s * F + c);
  float* ap = agg + (size_t)d * F + c;
  atomicAdd(ap + 0, v.x); atomicAdd(ap + 1, v.y);
  atomicAdd(ap + 2, v.z); atomicAdd(ap + 3, v.w);
}

// ---------------------------------------------------------------- fused WMMA GEMM
// out = act( rowscale(A1)@W1 [+ A2@W2] [+ rowscale(ADD)] [+ bias] )
// rowscale(p)[r] = p[r] * 1/max(cnt[r],1)   (segment-mean divide folded in)
// All shapes compile-time: k-loops fully unroll into immediate-offset
// ds_load / global_load feeding v_wmma_f32_16x16x4_f32.
// 4 waves/block, one 16x16 tile per wave.
template <int K1, int K2, int FO, bool RS1, bool HAS_A2, bool HAS_ADD, bool RSADD,
          bool BIAS, bool ACT>
__global__ __launch_bounds__(128) void wmma_gemm_t(
    const float* __restrict__ A1, const float* __restrict__ W1,
    const float* __restrict__ rs1,
    const float* __restrict__ A2, const float* __restrict__ W2,
    const float* __restrict__ ADD, const float* __restrict__ rsAdd,
    const float* __restrict__ bias,
    int M, float* __restrict__ out)
{
  constexpr int KT = K1 + K2;
  constexpr int RSTRIDE = KT + 4;           // pad: no LDS bank conflicts on row reads
  __shared__ float smem[4 * 16 * RSTRIDE];

  const int lane = threadIdx.x & 31;
  const int wv   = threadIdx.x >> 5;
  const int tilesM = M >> 4;
  int mtile = blockIdx.x * 4 + wv;
  if (mtile >= tilesM) mtile = tilesM - 1;  // clamp: dup waves write identical values
  const int m0 = mtile << 4;
  const int n0 = blockIdx.y << 4;

  float* sA = smem + wv * 16 * RSTRIDE;

  // stage A1 (with mean-divide) and optional A2 tile into LDS, coalesced
#pragma unroll
  for (int idx = lane; idx < 16 * K1; idx += 32) {
    int r = idx / K1, c = idx - r * K1;
    float v = A1[(unsigned)(m0 + r) * K1 + c];
    if (RS1) v *= 1.0f / fmaxf(rs1[m0 + r], 1.0f);
    sA[r * RSTRIDE + c] = v;
  }
  if (HAS_A2) {
#pragma unroll
    for (int idx = lane; idx < 16 * K2; idx += 32) {
      int r = idx / K2, c = idx - r * K2;
      sA[r * RSTRIDE + K1 + c] = A2[(unsigned)(m0 + r) * K2 + c];
    }
  }
  __syncthreads();

  const int hh   = lane >> 4;               // half-wave: K pair select / M half
  const int rr   = lane & 15;
  const int col  = n0 + rr;
  const int koff = 2 * hh;

  const float* arow = sA + rr * RSTRIDE + koff;     // ds: imm offsets k, k+1
  const float* w1p  = W1 + (unsigned)koff * FO + col; // global: imm offset k*FO*4

  v8f acc = {};
#pragma unroll
  for (int k = 0; k < K1; k += 4) {
    v2f a; a.x = arow[k]; a.y = arow[k + 1];
    v2f b; b.x = w1p[k * FO]; b.y = w1p[k * FO + FO];
    acc = __builtin_amdgcn_wmma_f32_16x16x4_f32(
        false, a, false, b, (short)0, acc, false, false);
  }
  if (HAS_A2) {
    const float* arow2 = arow + K1;
    const float* w2p   = W2 + (unsigned)koff * FO + col;
#pragma unroll
    for (int k = 0; k < K2; k += 4) {
      v2f a; a.x = arow2[k]; a.y = arow2[k + 1];
      v2f b; b.x = w2p[k * FO]; b.y = w2p[k * FO + FO];
      acc = __builtin_amdgcn_wmma_f32_16x16x4_f32(
          false, a, false, b, (short)0, acc, false, false);
    }
  }

#pragma unroll
  for (int v = 0; v < 8; ++v) {
    int row = m0 + v + 8 * hh;
    unsigned o = (unsigned)row * FO + col;
    float val = acc[v];
    if (HAS_ADD) {
      float av = ADD[o];
      if (RSADD) av *= 1.0f / fmaxf(rsAdd[row], 1.0f);
      val += av;
    }
    if (BIAS) val += bias[col];
    if (ACT)  val = fmaxf(val, 0.0f);
    out[o] = val;
  }
}

// ---------------------------------------------------------------- batchnorm
__global__ void colstats_kernel(const float* __restrict__ h, long N, int Fo,
                                int rowsPerBlock,
                                float* __restrict__ sum, float* __restrict__ sumsq) {
  __shared__ float s1[256];
  __shared__ float s2[256];
  int tid = threadIdx.x;
  int c = tid % Fo, sub = tid / Fo, tpc = 256 / Fo;
  long r0 = (long)blockIdx.x * rowsPerBlock;
  long rend = r0 + rowsPerBlock; if (rend > N) rend = N;
  float a = 0.0f, b = 0.0f;
  for (long r = r0 + sub; r < rend; r += tpc) {
    float v = h[r * Fo + c]; a += v; b += v * v;
  }
  s1[tid] = a; s2[tid] = b;
  __syncthreads();
  for (int s = tpc >> 1; s > 0; s >>= 1) {
    if (sub < s) { s1[tid] += s1[tid + s * Fo]; s2[tid] += s2[tid + s * Fo]; }
    __syncthreads();
  }
  if (sub == 0) { atomicAdd(&sum[c], s1[tid]); atomicAdd(&sumsq[c], s2[tid]); }
}

__global__ void bn_apply_kernel(float* __restrict__ h, long N, int Fo,
                                const float* __restrict__ sum,
                                const float* __restrict__ sumsq,
                                const float* __restrict__ g,
                                const float* __restrict__ be) {
  long total = N * Fo;
  float invN = 1.0f / (float)N;
  long i = (long)blockIdx.x * blockDim.x + threadIdx.x;
  long stride = (long)gridDim.x * blockDim.x;
  for (; i < total; i += stride) {
    int c = (int)(i % Fo);
    float mu = sum[c] * invN;
    float var = sumsq[c] * invN - mu * mu;
    h[i] = (h[i] - mu) * rsqrtf(var + BN_EPS) * g[c] + be[c];
  }
}

// ---------------------------------------------------------------- pooling + head
__global__ void pool_kernel(const float* __restrict__ h, const int* __restrict__ batch,
                            long N, float* __restrict__ pooled) {
  long i = (long)blockIdx.x * blockDim.x + threadIdx.x;
  if (i >= N * 32) return;
  long n = i >> 5; int f = (int)(i & 31);
  atomicAdd(&pooled[(size_t)batch[n] * 32 + f], h[i]);
}

__global__ void final_kernel(const float* __restrict__ cr2,
                             const float* __restrict__ Wf3,
                             const float* __restrict__ bf3,
                             float* __restrict__ out) {
  int t = blockIdx.x * blockDim.x + threadIdx.x;
  if (t >= N_GRAPHS * 2) return;
  int g = t >> 1, j = t & 1;
  float s = bf3[j];
#pragma unroll
  for (int k = 0; k < 64; ++k) s += cr2[g * 64 + k] * Wf3[k * 2 + j];
  out[t] = s;
}

// ---------------------------------------------------------------- driver
extern "C" void kernel_launch(void* const* d_in, const int* in_sizes, int n_in,
                              void* d_out, int out_size, void* d_ws, size_t ws_size,
                              hipStream_t stream) {
  (void)n_in; (void)out_size; (void)ws_size;
  const float* x     = (const float*)d_in[0];
  const int*   ei    = (const int*)d_in[1];
  const int*   batch = (const int*)d_in[2];
  const long N = in_sizes[0] / 64;
  const long E = in_sizes[1] / 2;
  const int* src = ei;
  const int* dst = ei + E;

  const float *W1l = (const float*)d_in[3],  *b1 = (const float*)d_in[4],
              *W1r = (const float*)d_in[5],  *g1 = (const float*)d_in[6],  *be1 = (const float*)d_in[7];
  const float *W2l = (const float*)d_in[8],  *b2 = (const float*)d_in[9],
              *W2r = (const float*)d_in[10], *g2 = (const float*)d_in[11], *be2 = (const float*)d_in[12];
  const float *W3l = (const float*)d_in[13], *b3 = (const float*)d_in[14],
              *W3r = (const float*)d_in[15], *g3 = (const float*)d_in[16], *be3 = (const float*)d_in[17];
  const float *Wf1 = (const float*)d_in[18], *bf1 = (const float*)d_in[19];
  const float *Wf2 = (const float*)d_in[20], *bf2 = (const float*)d_in[21];
  const float *Wf3 = (const float*)d_in[22], *bf3 = (const float*)d_in[23];

  // workspace layout
  float* ws     = (float*)d_ws;
  float* cnt    = ws;                      // N
  float* agg    = cnt + N;                 // N*64
  float* T      = agg + N * 64;            // N*64
  float* h1     = T + N * 64;              // N*128
  float* h2     = h1 + N * 128;            // N*64
  float* h3     = h2 + N * 64;             // N*32
  float* ssum   = h3 + N * 32;             // 128
  float* ssq    = ssum + 128;              // 128
  float* pooled = ssq + 128;               // 512*32
  float* cr1    = pooled + N_GRAPHS * 32;  // 512*128
  float* cr2    = cr1 + N_GRAPHS * 128;    // 512*64

  auto zero = [&](float* p, long n) {
    long b = (n + 255) / 256; if (b > 4096) b = 4096;
    zero_kernel<<<(int)b, 256, 0, stream>>>(p, n);
  };
  auto bn = [&](float* h, int Fo, const float* g, const float* be) {
    zero(ssum, 256);
    colstats_kernel<<<(int)((N + 511) / 512), 256, 0, stream>>>(h, N, Fo, 512, ssum, ssq);
    long total = N * Fo; long b = (total + 255) / 256; if (b > 8192) b = 8192;
    bn_apply_kernel<<<(int)b, 256, 0, stream>>>(h, N, Fo, ssum, ssq, g, be);
  };
  auto ggrid = [](int M, int Fo) { return dim3((M / 16 + 3) / 4, Fo / 16); };

  // degrees (shared by all layers)
  zero(cnt, N);
  degree_kernel<<<(int)((E + 255) / 256), 256, 0, stream>>>(dst, E, cnt);

  // ---- layer 1: aggregate x (64-dim, the smaller side); dual-GEMM fuse
  zero(agg, N * 64);
  scatter_add_kernel<16><<<(int)((E * 16 + 255) / 256), 256, 0, stream>>>(x, src, dst, E, agg);
  wmma_gemm_t<64, 64, 128, true, true, false, false, true, true>
      <<<ggrid((int)N, 128), 128, 0, stream>>>(agg, W1l, cnt, x, W1r,
                                               nullptr, nullptr, b1, (int)N, h1);
  bn(h1, 128, g1, be1);

  // ---- layer 2: pre-transform to 64-dim before aggregating (matmul commutes with sum)
  wmma_gemm_t<128, 0, 64, false, false, false, false, false, false>
      <<<ggrid((int)N, 64), 128, 0, stream>>>(h1, W2l, nullptr, nullptr, nullptr,
                                              nullptr, nullptr, nullptr, (int)N, T);
  zero(agg, N * 64);
  scatter_add_kernel<16><<<(int)((E * 16 + 255) / 256), 256, 0, stream>>>(T, src, dst, E, agg);
  wmma_gemm_t<128, 0, 64, false, false, true, true, true, true>
      <<<ggrid((int)N, 64), 128, 0, stream>>>(h1, W2r, nullptr, nullptr, nullptr,
                                              agg, cnt, b2, (int)N, h2);
  bn(h2, 64, g2, be2);

  // ---- layer 3: pre-transform to 32-dim before aggregating
  wmma_gemm_t<64, 0, 32, false, false, false, false, false, false>
      <<<ggrid((int)N, 32), 128, 0, stream>>>(h2, W3l, nullptr, nullptr, nullptr,
                                              nullptr, nullptr, nullptr, (int)N, T);
  zero(agg, N * 32);
  scatter_add_kernel<8><<<(int)((E * 8 + 255) / 256), 256, 0, stream>>>(T, src, dst, E, agg);
  wmma_gemm_t<64, 0, 32, false, false, true, true, true, true>
      <<<ggrid((int)N, 32), 128, 0, stream>>>(h2, W3r, nullptr, nullptr, nullptr,
                                              agg, cnt, b3, (int)N, h3);
  bn(h3, 32, g3, be3);

  // ---- pooling + MLP head
  zero(pooled, N_GRAPHS * 32);
  pool_kernel<<<(int)((N * 32 + 255) / 256), 256, 0, stream>>>(h3, batch, N, pooled);
  wmma_gemm_t<32, 0, 128, false, false, false, false, true, true>
      <<<ggrid(N_GRAPHS, 128), 128, 0, stream>>>(pooled, Wf1, nullptr, nullptr, nullptr,
                                                 nullptr, nullptr, bf1, N_GRAPHS, cr1);
  wmma_gemm_t<128, 0, 64, false, false, false, false, true, true>
      <<<ggrid(N_GRAPHS, 64), 128, 0, stream>>>(cr1, Wf2, nullptr, nullptr, nullptr,
                                                nullptr, nullptr, bf2, N_GRAPHS, cr2);
  final_kernel<<<4, 256, 0, stream>>>(cr2, Wf3, bf3, (float*)d_out);
}